// OccupancyGrid_5669356831505
// MI455X (gfx1250) — compile-verified
//
#include <hip/hip_runtime.h>
#include <hip/hip_bf16.h>

#define RES 256
#define NC (RES * RES * RES)       // 16,777,216 cells
#define OCC_THRE 0.01f
#define EMA 0.95f
#define NSAMP 4000000
#define A_BLOCKS (NC / (4 * 256))  // 16384 blocks, 4 cells/thread, 256 thr/blk

typedef __attribute__((ext_vector_type(2))) float v2f;
typedef __attribute__((ext_vector_type(8))) float v8f;

// Exact FP32 wave(32) sum using V_WMMA_F32_16X16X4_F32 with all-ones A.
// B holds one value per lane in its first VGPR (second VGPR = 0), so
// D[m][n] = sum_k B[k][n]; column sums collect each lane's value exactly
// once regardless of the precise (K,N) lane mapping. Rows of D are equal,
// so a 16-wide xor-shuffle tree over c[0] yields the full 32-lane total
// in every lane. Requires EXEC == all ones (guaranteed at call sites).
__device__ __forceinline__ float wave_sum32(float s) {
    v2f a; a[0] = 1.0f; a[1] = 1.0f;   // all-ones A (layout-independent)
    v2f b; b[0] = s;    b[1] = 0.0f;
    v8f c = {};
    c = __builtin_amdgcn_wmma_f32_16x16x4_f32(
            /*neg_a=*/false, a, /*neg_b=*/false, b,
            /*c_mod=*/(short)0, c, /*reuse_a=*/false, /*reuse_b=*/false);
    float t = c[0];
    t += __shfl_xor(t, 8, 32);
    t += __shfl_xor(t, 4, 32);
    t += __shfl_xor(t, 2, 32);
    t += __shfl_xor(t, 1, 32);
    return t;
}

// Deterministic 256-thread block sum (8 waves on wave32).
__device__ __forceinline__ float block_sum256(float s, float* lds) {
    float w = wave_sum32(s);
    const int lane = threadIdx.x & 31;
    const int wid  = threadIdx.x >> 5;
    if (lane == 0) lds[wid] = w;
    __syncthreads();
    float t = 0.0f;
    if (threadIdx.x < 8) t = lds[threadIdx.x];
    t += __shfl_xor(t, 4, 32);
    t += __shfl_xor(t, 2, 32);
    t += __shfl_xor(t, 1, 32);
    return t;  // valid in thread 0
}

// Phase 1: occupancy EMA update + per-block partial sums.
// 4 cells per thread -> jitter reads become three aligned float4s.
__global__ void og_grid_update(const float* __restrict__ occs,
                               const float* __restrict__ jitter,
                               const float* __restrict__ roi,
                               float* __restrict__ occs_new,
                               float* __restrict__ partials) {
    __shared__ float lds[8];
    const float rx0 = roi[0], ry0 = roi[1], rz0 = roi[2];
    const float ex = roi[3] - rx0, ey = roi[4] - ry0, ez = roi[5] - rz0;

    const int t  = blockIdx.x * blockDim.x + threadIdx.x;
    const int i0 = t * 4;

    const float4* j4 = (const float4*)(jitter + (size_t)i0 * 3);
    const float4 ja = j4[0], jb = j4[1], jc = j4[2];
    const float4 oc = *(const float4*)(occs + i0);

    const float jx[4] = {ja.x, ja.w, jb.z, jc.y};
    const float jy[4] = {ja.y, jb.x, jb.w, jc.z};
    const float jz[4] = {ja.z, jb.y, jc.x, jc.w};
    const float oo[4] = {oc.x, oc.y, oc.z, oc.w};

    float4 outv;
    float* op = &outv.x;
    float s = 0.0f;
    const float inv = 1.0f / (float)RES;

#pragma unroll
    for (int k = 0; k < 4; ++k) {
        const int i = i0 + k;
        // meshgrid(indexing='ij'): i = (gx*256 + gy)*256 + gz
        const float gz = (float)(i & 255);
        const float gy = (float)((i >> 8) & 255);
        const float gx = (float)((i >> 16) & 255);
        const float px = (gx + jx[k]) * inv * ex + rx0;
        const float py = (gy + jy[k]) * inv * ey + ry0;
        const float pz = (gz + jz[k]) * inv * ez + rz0;
        const float occ = __expf(-0.5f * (px * px + py * py + pz * pz));
        const float v = fmaxf(oo[k] * EMA, occ);
        op[k] = v;
        s += v;
    }
    *(float4*)(occs_new + i0) = outv;

    const float bs = block_sum256(s, lds);
    if (threadIdx.x == 0) partials[blockIdx.x] = bs;
}

// Phase 2: fold 16384 partials -> thresh = min(mean, 0.01). Deterministic.
__global__ void og_reduce_thresh(const float* __restrict__ partials,
                                 float* __restrict__ thresh) {
    __shared__ float lds[8];
    float s = 0.0f;
    for (int i = threadIdx.x; i < A_BLOCKS; i += 256) s += partials[i];
    const float tot = block_sum256(s, lds);
    if (threadIdx.x == 0) {
        const float mean = tot / (float)NC;
        *thresh = fminf(mean, OCC_THRE);
    }
}

// Phase 3: sample -> voxel gather. 4 samples/thread (three aligned float4s).
__global__ void og_sample_gather(const float* __restrict__ samples,
                                 const float* __restrict__ occs_new,
                                 const float* __restrict__ roi,
                                 const float* __restrict__ thresh_p,
                                 float* __restrict__ out) {
    const int t  = blockIdx.x * blockDim.x + threadIdx.x;
    const int s0 = t * 4;
    if (s0 >= NSAMP) return;

    const float rx0 = roi[0], ry0 = roi[1], rz0 = roi[2];
    const float sx = (float)RES / (roi[3] - rx0);
    const float sy = (float)RES / (roi[4] - ry0);
    const float sz = (float)RES / (roi[5] - rz0);
    const float th = *thresh_p;

    const float4* p4 = (const float4*)(samples + (size_t)s0 * 3);
    const float4 a = p4[0], b = p4[1], c = p4[2];
    const float X[4] = {a.x, a.w, b.z, c.y};
    const float Y[4] = {a.y, b.x, b.w, c.z};
    const float Z[4] = {a.z, b.y, c.x, c.w};

    float4 outv;
    float* op = &outv.x;
#pragma unroll
    for (int k = 0; k < 4; ++k) {
        const int ix = min(max((int)floorf((X[k] - rx0) * sx), 0), RES - 1);
        const int iy = min(max((int)floorf((Y[k] - ry0) * sy), 0), RES - 1);
        const int iz = min(max((int)floorf((Z[k] - rz0) * sz), 0), RES - 1);
        const int flat = (ix * RES + iy) * RES + iz;
        const float v = occs_new[flat];
        op[k] = (v > th) ? v : 0.0f;
    }
    *(float4*)(out + s0) = outv;
}

extern "C" void kernel_launch(void* const* d_in, const int* in_sizes, int n_in,
                              void* d_out, int out_size, void* d_ws, size_t ws_size,
                              hipStream_t stream) {
    const float* samples = (const float*)d_in[0];   // 4,000,000 x 3
    const float* occs    = (const float*)d_in[1];   // 16,777,216
    const float* jitter  = (const float*)d_in[2];   // 16,777,216 x 3
    const float* roi     = (const float*)d_in[3];   // 6
    float* out = (float*)d_out;                     // 4,000,000

    // Workspace layout (every word we read is written first; no memset needed):
    float* occs_new = (float*)d_ws;                 // NC floats (64 MB)
    float* partials = occs_new + NC;                // A_BLOCKS floats
    float* thresh   = partials + A_BLOCKS;          // 1 float

    og_grid_update<<<A_BLOCKS, 256, 0, stream>>>(occs, jitter, roi,
                                                 occs_new, partials);
    og_reduce_thresh<<<1, 256, 0, stream>>>(partials, thresh);

    const int nthreads = NSAMP / 4;                 // 1,000,000
    const int blocks   = (nthreads + 255) / 256;    // 3907
    og_sample_gather<<<blocks, 256, 0, stream>>>(samples, occs_new, roi,
                                                 thresh, out);
}